// GATSingleLayer_13280038879721
// MI455X (gfx1250) — compile-verified
//
#include <hip/hip_runtime.h>
#include <math.h>

typedef float v2f __attribute__((ext_vector_type(2)));
typedef float v8f __attribute__((ext_vector_type(8)));

#define FDIM   256   // F_IN == HEADS*F_HEAD == 256
#define HEADS  8
#define FHEAD  32
#define NEGSL  0.2f
#define KC     32    // K-chunk staged in LDS
#define WL_STRIDE 260   // 260 % 64 == 4  -> conflict-free column reads
#define XL_STRIDE 36    // 36*4B = 144B   -> 16B aligned rows, conflict-free

// ---------------------------------------------------------------------------
// Kernel 0: init out = bias (broadcast), nmax = -inf, denom = 0
// ---------------------------------------------------------------------------
__global__ __launch_bounds__(256) void gat_init(float* __restrict__ out,
                                                const float* __restrict__ bias,
                                                float* __restrict__ nmax,
                                                float* __restrict__ denom,
                                                int n_nodes) {
    int i = blockIdx.x * 256 + threadIdx.x;
    int total_out = n_nodes * FDIM;
    if (i < total_out) out[i] = bias[i & (FDIM - 1)];
    if (i < n_nodes * HEADS) {
        nmax[i]  = -INFINITY;
        denom[i] = 0.0f;
    }
}

// ---------------------------------------------------------------------------
// Kernel 1: h = x @ W  via V_WMMA_F32_16X16X4_F32, W/x staged through LDS
// grid.x = n_nodes/16 ; block = 256 (8 waves). Wave w owns cols [w*32, w*32+32).
// ---------------------------------------------------------------------------
__global__ __launch_bounds__(256) void gat_gemm_wmma(const float* __restrict__ x,
                                                     const float* __restrict__ W,
                                                     float* __restrict__ hbuf,
                                                     int n_nodes) {
    __shared__ float Wl[KC * WL_STRIDE]; // 33,280 B
    __shared__ float Xl[16 * XL_STRIDE]; //  2,304 B

    const int tid  = threadIdx.x;
    const int lane = tid & 31;
    const int wave = tid >> 5;           // 0..7
    const int m0   = blockIdx.x * 16;    // 16 rows per block (50000 % 16 == 0)
    const int n0   = wave * 32;          // two 16-col tiles per wave

    v8f c0 = {};
    v8f c1 = {};

    for (int k0 = 0; k0 < FDIM; k0 += KC) {
        // --- stage W[k0..k0+31][0..255] -> LDS (8 float4 per thread) ---
#pragma unroll
        for (int j = 0; j < 8; ++j) {
            int linear = tid + j * 256;        // float4 index 0..2047
            int row    = linear >> 6;          // 0..31
            int col4   = (linear & 63) << 2;   // 0..252
            float4 w4 = *(const float4*)(W + (size_t)(k0 + row) * FDIM + col4);
            *(float4*)(Wl + row * WL_STRIDE + col4) = w4;
        }
        // --- stage x[m0..m0+15][k0..k0+31] -> LDS (1 float4, 128 threads) ---
        if (tid < 128) {
            int row  = tid >> 3;               // 0..15
            int col4 = (tid & 7) << 2;         // 0..28
            float4 x4 = *(const float4*)(x + (size_t)(m0 + row) * FDIM + k0 + col4);
            *(float4*)(Xl + row * XL_STRIDE + col4) = x4;
        }
        __syncthreads();

        const int mrow = lane & 15;
        const int koff = (lane >> 4) * 2;      // 16x4 f32 A layout: K{0,1}|K{2,3}
        const int ncol = lane & 15;
#pragma unroll
        for (int kk = 0; kk < KC; kk += 4) {
            v2f a;
            a.x = Xl[mrow * XL_STRIDE + kk + koff];
            a.y = Xl[mrow * XL_STRIDE + kk + koff + 1];
            v2f b0, b1;
            b0.x = Wl[(kk + koff)     * WL_STRIDE + n0 + ncol];
            b0.y = Wl[(kk + koff + 1) * WL_STRIDE + n0 + ncol];
            b1.x = Wl[(kk + koff)     * WL_STRIDE + n0 + 16 + ncol];
            b1.y = Wl[(kk + koff + 1) * WL_STRIDE + n0 + 16 + ncol];
            c0 = __builtin_amdgcn_wmma_f32_16x16x4_f32(false, a, false, b0,
                                                       (short)0, c0, false, false);
            c1 = __builtin_amdgcn_wmma_f32_16x16x4_f32(false, a, false, b1,
                                                       (short)0, c1, false, false);
        }
        __syncthreads();
    }

    // C layout: VGPR r holds (M = r | r+8 by lane half, N = lane&15)
    const int crow = (lane >> 4) * 8;
    const int ccol = lane & 15;
#pragma unroll
    for (int r = 0; r < 8; ++r) {
        hbuf[(size_t)(m0 + crow + r) * FDIM + n0 + ccol]      = c0[r];
        hbuf[(size_t)(m0 + crow + r) * FDIM + n0 + 16 + ccol] = c1[r];
    }
}

// ---------------------------------------------------------------------------
// Kernel 2: a_src[n,h] = <h[n,h,:], att_src[h,:]> ; same for a_dst
// one thread per (node, head)
// ---------------------------------------------------------------------------
__global__ __launch_bounds__(256) void gat_attn(const float* __restrict__ hbuf,
                                                const float* __restrict__ att_src,
                                                const float* __restrict__ att_dst,
                                                float* __restrict__ asrc,
                                                float* __restrict__ adst,
                                                int n_nodes) {
    int i = blockIdx.x * 256 + threadIdx.x;
    if (i >= n_nodes * HEADS) return;
    int n  = i >> 3;
    int hd = i & 7;
    const float* hp = hbuf + (size_t)n * FDIM + hd * FHEAD;
    const float* as = att_src + hd * FHEAD;
    const float* ad = att_dst + hd * FHEAD;
    float s = 0.0f, d = 0.0f;
#pragma unroll 8
    for (int f = 0; f < FHEAD; ++f) {
        float v = hp[f];
        s = fmaf(v, as[f], s);
        d = fmaf(v, ad[f], d);
    }
    asrc[i] = s;
    adst[i] = d;
}

// Float atomic max via sign-aware integer trick (nmax initialized to -inf)
__device__ __forceinline__ void atomicMaxFloat(float* addr, float v) {
    if (v >= 0.0f)
        atomicMax((int*)addr, __float_as_int(v));
    else
        atomicMin((unsigned int*)addr, __float_as_uint(v));
}

__device__ __forceinline__ float leaky(float v) {
    return v > 0.0f ? v : NEGSL * v;
}

// ---------------------------------------------------------------------------
// Kernel 3: segment max of e over dst  (one thread per (edge, head))
// ---------------------------------------------------------------------------
__global__ __launch_bounds__(256) void gat_edge_max(const float* __restrict__ asrc,
                                                    const float* __restrict__ adst,
                                                    const int* __restrict__ srcI,
                                                    const int* __restrict__ dstI,
                                                    float* __restrict__ nmax,
                                                    int n_edges) {
    int i = blockIdx.x * 256 + threadIdx.x;
    if (i >= n_edges * HEADS) return;
    int e  = i >> 3;
    int hd = i & 7;
    int s = srcI[e], d = dstI[e];
    float v = leaky(asrc[s * HEADS + hd] + adst[d * HEADS + hd]);
    atomicMaxFloat(&nmax[d * HEADS + hd], v);
}

// ---------------------------------------------------------------------------
// Kernel 4: denom[d,h] += exp(e - max[d,h])
// ---------------------------------------------------------------------------
__global__ __launch_bounds__(256) void gat_edge_sum(const float* __restrict__ asrc,
                                                    const float* __restrict__ adst,
                                                    const int* __restrict__ srcI,
                                                    const int* __restrict__ dstI,
                                                    const float* __restrict__ nmax,
                                                    float* __restrict__ denom,
                                                    int n_edges) {
    int i = blockIdx.x * 256 + threadIdx.x;
    if (i >= n_edges * HEADS) return;
    int e  = i >> 3;
    int hd = i & 7;
    int s = srcI[e], d = dstI[e];
    float v = leaky(asrc[s * HEADS + hd] + adst[d * HEADS + hd]);
    atomicAdd(&denom[d * HEADS + hd], expf(v - nmax[d * HEADS + hd]));
}

// ---------------------------------------------------------------------------
// Kernel 5: out[dst] += alpha * h[src]  (one wave per edge, 8 floats/lane,
// lane's 8 features lie within exactly one head: head = lane>>2)
// ---------------------------------------------------------------------------
__global__ __launch_bounds__(256) void gat_aggregate(const float* __restrict__ hbuf,
                                                     const float* __restrict__ asrc,
                                                     const float* __restrict__ adst,
                                                     const float* __restrict__ nmax,
                                                     const float* __restrict__ denom,
                                                     const int* __restrict__ srcI,
                                                     const int* __restrict__ dstI,
                                                     float* __restrict__ out,
                                                     int n_edges) {
    int edge = blockIdx.x * 8 + (threadIdx.x >> 5);
    if (edge >= n_edges) return;
    int lane = threadIdx.x & 31;
    int s = srcI[edge], d = dstI[edge];
    int hd = lane >> 2;                         // 8 feats/lane -> one head/lane
    float v = leaky(asrc[s * HEADS + hd] + adst[d * HEADS + hd]);
    float alpha = expf(v - nmax[d * HEADS + hd]) / denom[d * HEADS + hd];
    const float* hs = hbuf + (size_t)s * FDIM + lane * 8;
    float*       op = out  + (size_t)d * FDIM + lane * 8;
#pragma unroll
    for (int j = 0; j < 8; ++j)
        atomicAdd(&op[j], alpha * hs[j]);
}

// ---------------------------------------------------------------------------
extern "C" void kernel_launch(void* const* d_in, const int* in_sizes, int n_in,
                              void* d_out, int out_size, void* d_ws, size_t ws_size,
                              hipStream_t stream) {
    const float* x       = (const float*)d_in[0];
    const float* W       = (const float*)d_in[1];
    const float* att_src = (const float*)d_in[2];
    const float* att_dst = (const float*)d_in[3];
    const float* bias    = (const float*)d_in[4];
    const int*   edge    = (const int*)d_in[5];

    const int n_nodes = in_sizes[0] / FDIM;
    const int n_edges = in_sizes[5] / 2;
    const int* srcI = edge;
    const int* dstI = edge + n_edges;

    float* hbuf  = (float*)d_ws;                          // n_nodes*256 f32
    float* asrc  = hbuf + (size_t)n_nodes * FDIM;         // n_nodes*8
    float* adst  = asrc + (size_t)n_nodes * HEADS;
    float* nmax  = adst + (size_t)n_nodes * HEADS;
    float* denom = nmax + (size_t)n_nodes * HEADS;
    float* out   = (float*)d_out;

    {   // init out/bias + per-node softmax state
        int total = n_nodes * FDIM;
        gat_init<<<(total + 255) / 256, 256, 0, stream>>>(out, bias, nmax, denom, n_nodes);
    }
    // h = x @ W  (WMMA f32)
    gat_gemm_wmma<<<n_nodes / 16, 256, 0, stream>>>(x, W, hbuf, n_nodes);
    {   // per-node attention coefficients
        int total = n_nodes * HEADS;
        gat_attn<<<(total + 255) / 256, 256, 0, stream>>>(hbuf, att_src, att_dst,
                                                          asrc, adst, n_nodes);
    }
    {   // segment softmax over incoming edges
        int total = n_edges * HEADS;
        gat_edge_max<<<(total + 255) / 256, 256, 0, stream>>>(asrc, adst, srcI, dstI,
                                                              nmax, n_edges);
        gat_edge_sum<<<(total + 255) / 256, 256, 0, stream>>>(asrc, adst, srcI, dstI,
                                                              nmax, denom, n_edges);
    }
    // attention-weighted scatter aggregation
    gat_aggregate<<<(n_edges + 7) / 8, 256, 0, stream>>>(hbuf, asrc, adst, nmax, denom,
                                                         srcI, dstI, out, n_edges);
}